// NPMT_43525198577922
// MI455X (gfx1250) — compile-verified
//
#include <hip/hip_runtime.h>
#include <math.h>

// Problem constants (match reference)
#define Tt   128
#define Bb   8
#define Dd   256
#define Vv   10000
#define MSs  4
#define Ss   (MSs + 1)
#define WINw 1
#define EOSi 4
#define LINF 1000000.0f

typedef __attribute__((ext_vector_type(2))) float v2f;
typedef __attribute__((ext_vector_type(8))) float v8f;

__device__ __forceinline__ float sigmf(float x) { return 1.0f / (1.0f + expf(-x)); }

// Issue one 16-byte async global->LDS copy (ASYNCcnt-tracked, gfx1250).
__device__ __forceinline__ void async_g2l_b128(void* lds_ptr, const void* gptr)
{
  unsigned lds_off = (unsigned)(size_t)lds_ptr;   // generic->LDS: low 32 bits
  asm volatile("global_load_async_to_lds_b128 %0, %1, off"
               :: "v"(lds_off), "v"(gptr) : "memory");
}
__device__ __forceinline__ void wait_async0()
{
  asm volatile("s_wait_asynccnt 0" ::: "memory");
}

// ---------------------------------------------------------------------------
// Generic fp32 WMMA GEMM (K fixed = 256):
//   C[M,N] (+)= a_scale * A[M,256] @ W[N,256]^T (+bias) (tanh?)
// The block's 16-row A panel is staged once into LDS via async-to-LDS and
// shared by all 4 waves; each wave owns one 16x16 N tile, WMMA f32 16x16x4
// chained over K. grid = (M/16, N/64), block = 128.
// ---------------------------------------------------------------------------
__global__ __launch_bounds__(128) void wmma_gemm_awt(
    const float* __restrict__ A, int lda,
    const float* __restrict__ W,
    const float* __restrict__ bias,
    float* __restrict__ C, int ldc,
    float a_scale, int acc, int act)
{
  __shared__ __align__(16) float As[16][260];
  const int tid  = threadIdx.x;
  const int lane = tid & 31;
  const int wave = tid >> 5;
  const int m0 = blockIdx.x << 4;
  const int n0 = (blockIdx.y << 6) + (wave << 4);
  const int lm = lane & 15;             // M (A) / N (B) index within tile
  const int lk = (lane >> 4) << 1;      // K sub-offset for this lane half

  // stage A panel: 16 rows x 256 f32 = 1024 16B chunks, 8 per thread
  for (int cch = tid; cch < 16 * 64; cch += 128) {
    int row = cch >> 6, col = cch & 63;
    async_g2l_b128(&As[row][col * 4], A + (size_t)(m0 + row) * lda + col * 4);
  }
  wait_async0();
  __syncthreads();

  const float* wrow = W + (size_t)(n0 + lm) * Dd;
  v8f d = {};
  for (int k = 0; k < Dd; k += 4) {
    v2f a, b;
    a[0] = As[lm][k + lk]; a[1] = As[lm][k + lk + 1];
    b[0] = wrow[k + lk];   b[1] = wrow[k + lk + 1];
    d = __builtin_amdgcn_wmma_f32_16x16x4_f32(false, a, false, b, (short)0, d, false, false);
  }
  const int mb = m0 + ((lane >> 4) << 3);   // D frag: M = r + 8*(lane/16)
  const int n  = n0 + lm;
  const float bv = bias ? bias[n] : 0.0f;
#pragma unroll
  for (int r = 0; r < 8; ++r) {
    float v = a_scale * d[r] + bv;
    float* cp = C + (size_t)(mb + r) * ldc + n;
    if (acc) v += *cp;
    if (act) v = tanhf(v);
    *cp = v;
  }
}

// ---------------------------------------------------------------------------
// x[b,t,:] = emb[x_ids[b, WIN+t]]       (B,T,D)
// ---------------------------------------------------------------------------
__global__ void k_gather_x(const int* __restrict__ ids, const float* __restrict__ emb,
                           float* __restrict__ x)
{
  int e = blockIdx.x * blockDim.x + threadIdx.x;      // B*T*D
  int d = e & (Dd - 1);
  int bt = e >> 8;
  int t = bt % Tt, b = bt / Tt;
  int tok = ids[b * (Tt + 2 * WINw) + WINw + t];
  x[e] = emb[(size_t)tok * Dd + d];
}

// seqX[(t*B+b),d] = t==0 ? y_start[d] : x[b,t-1,d]
__global__ void k_seqx(const float* __restrict__ x, const float* __restrict__ y_start,
                       float* __restrict__ seqX)
{
  int e = blockIdx.x * blockDim.x + threadIdx.x;      // T*B*D
  int d = e & (Dd - 1);
  int r = e >> 8;
  int t = r / Bb, b = r % Bb;
  seqX[e] = (t == 0) ? y_start[d] : x[((size_t)b * Tt + (t - 1)) * Dd + d];
}

// ---------------------------------------------------------------------------
// Persistent single-workgroup encoder LSTM scan (recurrence is latency bound,
// B=8 only). h lives in LDS (padded to 16 rows of zeros for the WMMA M-tile);
// per step: g = h @ Wh^T via WMMA (input projection gx_all precomputed).
// ---------------------------------------------------------------------------
__global__ __launch_bounds__(256) void k_enc_scan(
    const float* __restrict__ gx_all,   // (T*B, 4D), includes cell_b
    const float* __restrict__ Wh,       // (4D, D)
    float* __restrict__ h_seq)          // (T*B, D)
{
  __shared__ float hbuf[16][Dd + 8];    // rows 8..15 stay zero (tile padding)
  __shared__ float cbuf[Bb][Dd];
  __shared__ float gbuf[Bb][4 * Dd];
  const int tid = threadIdx.x;
  const int lane = tid & 31, wave = tid >> 5;
  const int lm = lane & 15, lk = (lane >> 4) << 1;
  for (int i = tid; i < 16 * (Dd + 8); i += 256) ((float*)hbuf)[i] = 0.0f;
  for (int i = tid; i < Bb * Dd; i += 256) ((float*)cbuf)[i] = 0.0f;
  __syncthreads();

  for (int t = 0; t < Tt; ++t) {
    // g = h @ Wh^T : 64 N-tiles, 8 per wave
    for (int tile = wave; tile < 64; tile += 8) {
      const int n0 = tile << 4;
      const float* wrow = Wh + (size_t)(n0 + lm) * Dd;
      v8f d = {};
      for (int k = 0; k < Dd; k += 4) {
        v2f a, b;
        a[0] = hbuf[lm][k + lk]; a[1] = hbuf[lm][k + lk + 1];
        b[0] = wrow[k + lk];     b[1] = wrow[k + lk + 1];
        d = __builtin_amdgcn_wmma_f32_16x16x4_f32(false, a, false, b, (short)0, d, false, false);
      }
      if (lane < 16) {                      // rows 0..7 = real batch
#pragma unroll
        for (int r = 0; r < 8; ++r) gbuf[r][n0 + lm] = d[r];
      }
    }
    __syncthreads();
    const float* gx = gx_all + (size_t)t * Bb * 4 * Dd;
    for (int e = tid; e < Bb * Dd; e += 256) {
      int b = e >> 8, d = e & (Dd - 1);
      const float* gxr = gx + (size_t)b * 4 * Dd;
      float gi = gbuf[b][d]          + gxr[d];
      float gf = gbuf[b][Dd + d]     + gxr[Dd + d];
      float gg = gbuf[b][2 * Dd + d] + gxr[2 * Dd + d];
      float go = gbuf[b][3 * Dd + d] + gxr[3 * Dd + d];
      float c2 = sigmf(gf) * cbuf[b][d] + sigmf(gi) * tanhf(gg);
      float h2 = sigmf(go) * tanhf(c2);
      cbuf[b][d] = c2;
      hbuf[b][d] = h2;
      h_seq[((size_t)t * Bb + b) * Dd + d] = h2;
    }
    __syncthreads();
  }
}

// seg[((t*B+b)*S+s), d]: s==0 -> y_in0 row, else padded x window
__global__ void k_seg(const float* __restrict__ x, const float* __restrict__ y_in0,
                      float* __restrict__ seg)
{
  int e = blockIdx.x * blockDim.x + threadIdx.x;      // T*B*S*D
  int d = e & (Dd - 1);
  int rs = e >> 8;
  int s = rs % Ss, r = rs / Ss;
  int b = r % Bb, t = r / Bb;
  float v;
  if (s == 0) v = y_in0[(size_t)r * Dd + d];
  else {
    int tp = t + s - 1;
    v = (tp < Tt) ? x[((size_t)b * Tt + tp) * Dd + d] : 0.0f;
  }
  seg[e] = v;
}

// decoder init: h0/h1 from h0_all heads, c0=c1=0
__global__ void k_dec_init(const float* __restrict__ h0_all,
                           float* __restrict__ h0, float* __restrict__ c0,
                           float* __restrict__ h1, float* __restrict__ c1)
{
  int e = blockIdx.x * blockDim.x + threadIdx.x;      // (T*B)*D
  int r = e >> 8, d = e & (Dd - 1);
  h0[e] = h0_all[(size_t)r * 2 * Dd + d];
  h1[e] = h0_all[(size_t)r * 2 * Dd + Dd + d];
  c0[e] = 0.0f; c1[e] = 0.0f;
}

// LSTM gate elementwise: g has full preactivations; updates h,c; optional h copy-out
__global__ void k_gates(const float* __restrict__ g, int g_stride,
                        float* __restrict__ h, float* __restrict__ c,
                        float* __restrict__ hout, int hout_stride)
{
  int e = blockIdx.x * blockDim.x + threadIdx.x;      // rows*D
  int r = e >> 8, d = e & (Dd - 1);
  const float* gr = g + (size_t)r * g_stride;
  float gi = gr[d], gf = gr[Dd + d], gg = gr[2 * Dd + d], go = gr[3 * Dd + d];
  float c2 = sigmf(gf) * c[e] + sigmf(gi) * tanhf(gg);
  float h2 = sigmf(go) * tanhf(c2);
  c[e] = c2; h[e] = h2;
  if (hout) hout[(size_t)r * hout_stride + d] = h2;
}

// ---------------------------------------------------------------------------
// Fused logits GEMM + logsumexp over V (logits never materialized).
// Block = 128 threads (4 waves) covering 64 rows of dec_h (LDS-resident).
// All 4 waves walk the same 625 V-tiles and share one 16x256 emb tile that is
// double-buffered in LDS via global_load_async_to_lds_b128 (ASYNCcnt),
// overlapping the fetch of tile i+1 with the 64 chained WMMAs on tile i.
// Online max/sum per row slice, LDS combine -> logZ.
// ---------------------------------------------------------------------------
#define LZ_ROWS 64
__global__ __launch_bounds__(128) void k_logz(
    const float* __restrict__ dech, const float* __restrict__ emb,
    const float* __restrict__ out_b, float* __restrict__ logZ)
{
  __shared__ __align__(16) float As[LZ_ROWS][260];
  __shared__ __align__(16) float Bs[2][16 * 260];
  __shared__ float pm[4][32][8];
  __shared__ float ps[4][32][8];
  const int tid = threadIdx.x;
  const int lane = tid & 31, wave = tid >> 5;
  const int lm = lane & 15, lk = (lane >> 4) << 1;
  const int r0 = blockIdx.x * LZ_ROWS;

  // stage the block's 64-row A panel (async, 32 chunks per thread)
  for (int cch = tid; cch < LZ_ROWS * 64; cch += 128) {
    int row = cch >> 6, col = cch & 63;
    async_g2l_b128(&As[row][col * 4], dech + (size_t)(r0 + row) * Dd + col * 4);
  }
  // stage first emb tile into buffer 0
  for (int cch = tid; cch < 16 * 64; cch += 128) {
    int row = cch >> 6, col = cch & 63;
    async_g2l_b128(&Bs[0][row * 260 + col * 4], emb + (size_t)row * Dd + col * 4);
  }
  wait_async0();
  __syncthreads();

  float rmax[8], rsum[8];
#pragma unroll
  for (int r = 0; r < 8; ++r) { rmax[r] = -3.0e38f; rsum[r] = 0.0f; }

  const int arow = (wave << 4) + lm;      // this wave's A row within panel
  int p = 0;
  for (int vt = 0; vt < Vv / 16; ++vt) {
    // prefetch next tile into the other buffer (no wait yet)
    if (vt + 1 < Vv / 16) {
      const int v1 = (vt + 1) << 4;
      for (int cch = tid; cch < 16 * 64; cch += 128) {
        int row = cch >> 6, col = cch & 63;
        async_g2l_b128(&Bs[p ^ 1][row * 260 + col * 4],
                       emb + (size_t)(v1 + row) * Dd + col * 4);
      }
    }
    // compute on current buffer
    const int n0 = vt << 4;
    const float* brow = &Bs[p][lm * 260];
    v8f d = {};
    for (int k = 0; k < Dd; k += 4) {
      v2f a, b;
      a[0] = As[arow][k + lk]; a[1] = As[arow][k + lk + 1];
      b[0] = brow[k + lk];     b[1] = brow[k + lk + 1];
      d = __builtin_amdgcn_wmma_f32_16x16x4_f32(false, a, false, b, (short)0, d, false, false);
    }
    float ob = out_b[n0 + lm];
#pragma unroll
    for (int r = 0; r < 8; ++r) {
      float v = d[r] + ob;
      float m = fmaxf(rmax[r], v);
      rsum[r] = rsum[r] * expf(rmax[r] - m) + expf(v - m);
      rmax[r] = m;
    }
    wait_async0();
    __syncthreads();
    p ^= 1;
  }
#pragma unroll
  for (int r = 0; r < 8; ++r) { pm[wave][lane][r] = rmax[r]; ps[wave][lane][r] = rsum[r]; }
  __syncthreads();
  if (tid < 64) {                          // one thread per (wave, row)
    int w = tid >> 4, m = tid & 15;
    int lb = (m < 8) ? 0 : 16;
    int r = m & 7;
    float gm = -3.0e38f, gs = 0.0f;
    for (int l = lb; l < lb + 16; ++l) {
      float x = pm[w][l][r], s = ps[w][l][r];
      float nm = fmaxf(gm, x);
      gs = gs * expf(gm - nm) + s * expf(x - nm);
      gm = nm;
    }
    logZ[r0 + (w << 4) + m] = gm + logf(gs);
  }
}

// ---------------------------------------------------------------------------
// Target / EOS log-prob picks: one wave per dot product (256-wide) + shuffle reduce
// jobs 0..T*B*MS-1: tgt_lp ; jobs T*B*MS..2*T*B*MS-1: eos_lp
// ---------------------------------------------------------------------------
__global__ __launch_bounds__(256) void k_picks(
    const float* __restrict__ dech, const float* __restrict__ emb,
    const float* __restrict__ out_b, const float* __restrict__ logZ,
    const int* __restrict__ ids,
    float* __restrict__ tgt_lp, float* __restrict__ eos_lp)
{
  const int job = blockIdx.x * 8 + (threadIdx.x >> 5);
  const int lane = threadIdx.x & 31;
  const int half = Tt * Bb * MSs;
  const int which = (job >= half);
  const int j2 = which ? (job - half) : job;
  const int i = j2 % MSs;
  const int b = (j2 / MSs) % Bb;
  const int t = j2 / (MSs * Bb);
  int s, v;
  if (!which) {
    s = i;
    int tp = t + i; if (tp > Tt - 1) tp = Tt - 1;
    v = ids[b * (Tt + 2 * WINw) + WINw + tp];
  } else { s = i + 1; v = EOSi; }
  const int row = (t * Bb + b) * Ss + s;
  const float* a = dech + (size_t)row * Dd;
  const float* w = emb + (size_t)v * Dd;
  float p = 0.0f;
#pragma unroll
  for (int j = 0; j < Dd / 32; ++j) p += a[lane + 32 * j] * w[lane + 32 * j];
  for (int m = 16; m > 0; m >>= 1) p += __shfl_xor(p, m, 32);
  if (lane == 0) {
    float lp = p + out_b[v] - logZ[row];
    if (!which) tgt_lp[j2] = lp; else eos_lp[j2] = lp;
  }
}

// ---------------------------------------------------------------------------
// Final: cum/body/logpy in LDS, then the T-step DP (window MS) in LDS, scalar out.
// ---------------------------------------------------------------------------
__global__ __launch_bounds__(256) void k_final(
    const float* __restrict__ tgt_lp, const float* __restrict__ eos_lp,
    const int* __restrict__ lens, float* __restrict__ out)
{
  __shared__ float logpy[Tt * Ss * Bb];                // [t][m][b]
  __shared__ float Apad[MSs + Tt + 1][Bb];
  const int tid = threadIdx.x;
  for (int r = tid; r < Tt * Bb; r += 256) {
    int t = r / Bb, b = r % Bb;
    logpy[(t * Ss + 0) * Bb + b] = -LINF;
    float run = 0.0f;
    for (int i = 0; i < MSs; ++i) {
      if (t + i < Tt) run += tgt_lp[r * MSs + i];
      float body = (t + i + 1 <= Tt) ? (run + eos_lp[r * MSs + i]) : -LINF;
      logpy[(t * Ss + i + 1) * Bb + b] = body;
    }
  }
  for (int e = tid; e < (MSs + Tt + 1) * Bb; e += 256) ((float*)Apad)[e] = -LINF;
  __syncthreads();
  if (tid < Bb) Apad[MSs][tid] = 0.0f;
  __syncthreads();
  for (int j = 1; j <= Tt; ++j) {
    if (tid < Bb) {
      const int b = tid;
      float vals[MSs]; float mx = -3.0e38f;
      for (int i = 0; i < MSs; ++i) {
        int ss = j - MSs + i;
        float lpg = (ss >= 0) ? logpy[(ss * Ss + (MSs - i)) * Bb + b] : -LINF;
        float v = Apad[j + i][b] + lpg;
        vals[i] = v; mx = fmaxf(mx, v);
      }
      float sum = 0.0f;
      for (int i = 0; i < MSs; ++i) sum += expf(vals[i] - mx);
      Apad[MSs + j][b] = mx + logf(sum);
    }
    __syncthreads();
  }
  if (tid == 0) {
    float s = 0.0f; int sl = 0;
    for (int b = 0; b < Bb; ++b) { int L = lens[b]; sl += L; s += Apad[MSs + L][b]; }
    out[0] = -s / (float)sl;
  }
}

// ---------------------------------------------------------------------------
static inline void gemm(hipStream_t st, const float* A, int lda, const float* W,
                        const float* bias, float* C, int ldc,
                        int M, int N, float a_scale, int acc, int act)
{
  dim3 g(M / 16, N / 64), b(128);
  wmma_gemm_awt<<<g, b, 0, st>>>(A, lda, W, bias, C, ldc, a_scale, acc, act);
}

extern "C" void kernel_launch(void* const* d_in, const int* in_sizes, int n_in,
                              void* d_out, int out_size, void* d_ws, size_t ws_size,
                              hipStream_t stream)
{
  (void)in_sizes; (void)n_in; (void)out_size; (void)ws_size;
  const int*   x_ids   = (const int*)d_in[0];
  const int*   lens    = (const int*)d_in[1];
  const float* emb     = (const float*)d_in[2];
  const float* y_start = (const float*)d_in[3];
  const float* cell_Wi = (const float*)d_in[4];
  const float* cell_Wh = (const float*)d_in[5];
  const float* cell_b  = (const float*)d_in[6];
  const float* yin0_W  = (const float*)d_in[7];
  const float* yin0_b  = (const float*)d_in[8];
  const float* hlin_W  = (const float*)d_in[9];
  const float* hlin_b  = (const float*)d_in[10];
  const float* l0_Wi   = (const float*)d_in[11];
  const float* l0_Wh   = (const float*)d_in[12];
  const float* l0_b    = (const float*)d_in[13];
  const float* l1_Wi   = (const float*)d_in[14];
  const float* l1_Wh   = (const float*)d_in[15];
  const float* l1_b    = (const float*)d_in[16];
  const float* out_b   = (const float*)d_in[17];
  float* out = (float*)d_out;

  const int R  = Tt * Bb;          // 1024
  const int RS = R * Ss;           // 5120
  float* ws = (float*)d_ws;
  size_t o = 0;
  float* x      = ws + o; o += (size_t)Bb * Tt * Dd;      // 262144
  float* seqX   = ws + o; o += (size_t)R * Dd;            // 262144
  float* gx     = ws + o; o += (size_t)R * 4 * Dd;        // 1048576
  float* hseq   = ws + o; o += (size_t)R * Dd;
  float* yin0   = ws + o; o += (size_t)R * Dd;
  float* h0all  = ws + o; o += (size_t)R * 2 * Dd;
  float* seg    = ws + o; o += (size_t)RS * Dd;           // 1310720
  float* gi0    = ws + o; o += (size_t)RS * 4 * Dd;       // 5242880
  float* g1b    = ws + o; o += (size_t)R * 4 * Dd;
  float* h0     = ws + o; o += (size_t)R * Dd;
  float* c0     = ws + o; o += (size_t)R * Dd;
  float* h1     = ws + o; o += (size_t)R * Dd;
  float* c1     = ws + o; o += (size_t)R * Dd;
  float* dech   = ws + o; o += (size_t)RS * Dd;
  float* logZ   = ws + o; o += (size_t)RS;
  float* tlp    = ws + o; o += (size_t)R * MSs;
  float* elp    = ws + o; o += (size_t)R * MSs;

  // 1) embeddings + encoder input sequence
  k_gather_x<<<(Bb * Tt * Dd) / 256, 256, 0, stream>>>(x_ids, emb, x);
  k_seqx<<<(R * Dd) / 256, 256, 0, stream>>>(x, y_start, seqX);

  // 2) hoisted encoder input projection: gx = seqX @ cell_Wi^T + cell_b
  gemm(stream, seqX, Dd, cell_Wi, cell_b, gx, 4 * Dd, R, 4 * Dd, 1.0f, 0, 0);

  // 3) encoder LSTM scan (single persistent workgroup, h in LDS)
  k_enc_scan<<<1, 256, 0, stream>>>(gx, cell_Wh, hseq);

  // 4) projections of drop_h = 0.1*h_seq (scale folded into GEMM)
  gemm(stream, hseq, Dd, yin0_W, yin0_b, yin0, Dd, R, Dd, 0.1f, 0, 0);
  gemm(stream, hseq, Dd, hlin_W, hlin_b, h0all, 2 * Dd, R, 2 * Dd, 0.1f, 0, 1);

  // 5) decoder input segments + hoisted layer-0 input projection (all s at once)
  k_seg<<<(RS * Dd) / 256, 256, 0, stream>>>(x, yin0, seg);
  gemm(stream, seg, Dd, l0_Wi, l0_b, gi0, 4 * Dd, RS, 4 * Dd, 1.0f, 0, 0);

  // 6) decoder scan: batch T*B=1024 independent lanes, S=5 steps, 2 layers
  k_dec_init<<<(R * Dd) / 256, 256, 0, stream>>>(h0all, h0, c0, h1, c1);
  for (int s = 0; s < Ss; ++s) {
    // layer 0: gi0[:,s,:] += h0 @ l0_Wh^T ; gates
    gemm(stream, h0, Dd, l0_Wh, nullptr, gi0 + (size_t)s * 4 * Dd, Ss * 4 * Dd,
         R, 4 * Dd, 1.0f, 1, 0);
    k_gates<<<(R * Dd) / 256, 256, 0, stream>>>(gi0 + (size_t)s * 4 * Dd, Ss * 4 * Dd,
                                                h0, c0, nullptr, 0);
    // layer 1: g1b = h0 @ l1_Wi^T + l1_b ; g1b += h1 @ l1_Wh^T ; gates -> dec_h
    gemm(stream, h0, Dd, l1_Wi, l1_b, g1b, 4 * Dd, R, 4 * Dd, 1.0f, 0, 0);
    gemm(stream, h1, Dd, l1_Wh, nullptr, g1b, 4 * Dd, R, 4 * Dd, 1.0f, 1, 0);
    k_gates<<<(R * Dd) / 256, 256, 0, stream>>>(g1b, 4 * Dd, h1, c1,
                                                dech + (size_t)s * Dd, Ss * Dd);
  }

  // 7) fused logits GEMM + online logsumexp over V, async double-buffered emb tiles
  k_logz<<<RS / LZ_ROWS, 128, 0, stream>>>(dech, emb, out_b, logZ);

  // 8) target / EOS log-prob picks
  k_picks<<<(2 * R * MSs) / 8, 256, 0, stream>>>(dech, emb, out_b, logZ, x_ids, tlp, elp);

  // 9) DP + final scalar
  k_final<<<1, 256, 0, stream>>>(tlp, elp, lens, out);
}